// GraphEncoder_6390911336608
// MI455X (gfx1250) — compile-verified
//
#include <hip/hip_runtime.h>

typedef __attribute__((ext_vector_type(2))) float v2f;
typedef __attribute__((ext_vector_type(8))) float v8f;

#define DH 128  // hidden width, fixed by the model

// ---------------- utility: zero fill ----------------
__global__ void fill_zero(float* __restrict__ p, long n) {
  long i = (long)blockIdx.x * blockDim.x + threadIdx.x;
  long stride = (long)gridDim.x * blockDim.x;
  for (; i < n; i += stride) p[i] = 0.0f;
}

// ---------------- degrees (unweighted, 4 histograms in one pass) ----------------
__global__ void degree_kernel(const int* __restrict__ src_rc, const int* __restrict__ dst_rc,
                              const int* __restrict__ src_cr, const int* __restrict__ dst_cr,
                              float* __restrict__ degOutRow, float* __restrict__ degInCol,
                              float* __restrict__ degOutCol, float* __restrict__ degInRow,
                              int E) {
  int e = blockIdx.x * blockDim.x + threadIdx.x;
  if (e >= E) return;
  atomicAdd(&degOutRow[src_rc[e]], 1.0f);
  atomicAdd(&degInCol [dst_rc[e]], 1.0f);
  atomicAdd(&degOutCol[src_cr[e]], 1.0f);
  atomicAdd(&degInRow [dst_cr[e]], 1.0f);
}

__global__ void rsqrt_clamp_kernel(float* __restrict__ x, int n) {
  int i = blockIdx.x * blockDim.x + threadIdx.x;
  if (i < n) x[i] = rsqrtf(fmaxf(x[i], 1.0f));
}

// ---------------- fused gather * rsqrt(out_deg) * w_e -> scatter-add ----------------
// thread t handles edge e = t / (K/4), feature-quad v = t % (K/4)
__global__ void scatter_kernel(const float* __restrict__ h, const int* __restrict__ src,
                               const int* __restrict__ dst, const float* __restrict__ w,
                               const float* __restrict__ invOut, float* __restrict__ agg,
                               int E, int K) {
  int t = blockIdx.x * blockDim.x + threadIdx.x;
  int vecs = K >> 2;
  int e = t / vecs;
  int v = t - e * vecs;
  if (e >= E) return;
  int s = src[e], d = dst[e];
  float c = w[e] * invOut[s];
  float4 hv = ((const float4*)(h + (long)s * K))[v];
  float* ap = agg + (long)d * K + (v << 2);
  atomicAdd(ap + 0, hv.x * c);
  atomicAdd(ap + 1, hv.y * c);
  atomicAdd(ap + 2, hv.z * c);
  atomicAdd(ap + 3, hv.w * c);
}

// ---------------- WMMA GEMM: out = relu( (A * rsqrt(in_deg)[:,None]) @ W + b ) ----------------
// A: [N, K] (K = 64 or 128, compile-time), W: [K, 128] staged in LDS, out: [N, 128].
// Block = 256 threads = 8 waves; wave w owns output columns [16w, 16w+16);
// block b owns rows [16b, 16b+16). Fully unrolled K loop -> straight-line
// v_wmma_f32_16x16x4_f32 chain with clause-batched A loads.
// Fragment layouts per CDNA5 ISA 7.12.2:
//   A 16x4 f32 : lanes 0-15 -> M=lane (K = k..k+1 in the 2 VGPRs), lanes 16-31 -> K = k+2..k+3
//   B 4x16 f32 : lanes 0-15 -> N=lane (K = k..k+1),                lanes 16-31 -> K = k+2..k+3
//   C/D 16x16  : VGPR i, lanes 0-15 -> M=i, N=lane; lanes 16-31 -> M=i+8
template <int K>
__global__ __launch_bounds__(256) void gemm_norm_bias_relu(
    const float* __restrict__ A, const float* __restrict__ invIn,
    const float* __restrict__ W, const float* __restrict__ bias,
    float* __restrict__ out, int N) {
  extern __shared__ float sW[];  // K * 128 floats (<= 64 KB)
  for (int i = threadIdx.x; i < K * DH; i += 256) sW[i] = W[i];
  __syncthreads();

  const int wave = threadIdx.x >> 5;
  const int lane = threadIdx.x & 31;
  const int hi   = lane >> 4;    // 0: K-pair {k,k+1}, 1: K-pair {k+2,k+3}
  const int lm   = lane & 15;
  const int r0   = blockIdx.x * 16;
  if (r0 >= N) return;
  const int c0   = wave * 16;

  const int row = r0 + lm;
  const float s = invIn[row];
  const float2* Arow = (const float2*)(A + (long)row * K);

  constexpr int STEPS = K / 4;  // 16 or 32 WMMA ops

  // Prefetch all A fragments for this row (clause-batched loads), then scale.
  float2 afrag[STEPS];
#pragma unroll
  for (int i = 0; i < STEPS; ++i) afrag[i] = Arow[hi + 2 * i];
#pragma unroll
  for (int i = 0; i < STEPS; ++i) {
    afrag[i].x *= s;
    afrag[i].y *= s;
  }

  v8f acc = {};
#pragma unroll
  for (int i = 0; i < STEPS; ++i) {
    const int ka = (hi + 2 * i) << 1;  // absolute K index of this fragment's first col
    v2f a, b;
    a.x = afrag[i].x;
    a.y = afrag[i].y;
    b.x = sW[ka * DH + c0 + lm];
    b.y = sW[(ka + 1) * DH + c0 + lm];
    acc = __builtin_amdgcn_wmma_f32_16x16x4_f32(false, a, false, b, (short)0, acc,
                                                false, false);
  }

  const float bb = bias[c0 + lm];
#pragma unroll
  for (int i = 0; i < 8; ++i) {
    int m = r0 + (hi << 3) + i;
    float v = acc[i] + bb;
    out[(long)m * DH + c0 + lm] = v > 0.0f ? v : 0.0f;
  }
}

// ---------------- final: mean over rows of both matrices, summed ----------------
__global__ void colmean_kernel(const float* __restrict__ hrow, const float* __restrict__ hcol,
                               float* __restrict__ out, int nrow, int ncol) {
  __shared__ float red[256];
  int f = blockIdx.x;  // 0..127
  float a = 0.0f, b = 0.0f;
  for (int r = threadIdx.x; r < nrow; r += 256) a += hrow[(long)r * DH + f];
  for (int r = threadIdx.x; r < ncol; r += 256) b += hcol[(long)r * DH + f];
  red[threadIdx.x] = a / (float)nrow + b / (float)ncol;
  __syncthreads();
  for (int s = 128; s > 0; s >>= 1) {
    if (threadIdx.x < s) red[threadIdx.x] += red[threadIdx.x + s];
    __syncthreads();
  }
  if (threadIdx.x == 0) out[f] = red[0];
}

extern "C" void kernel_launch(void* const* d_in, const int* in_sizes, int n_in,
                              void* d_out, int out_size, void* d_ws, size_t ws_size,
                              hipStream_t stream) {
  const float* feat_row = (const float*)d_in[0];
  const float* feat_col = (const float*)d_in[1];
  const int*   src_rc   = (const int*)d_in[2];
  const int*   dst_rc   = (const int*)d_in[3];
  const float* w_rc     = (const float*)d_in[4];
  const int*   src_cr   = (const int*)d_in[5];
  const int*   dst_cr   = (const int*)d_in[6];
  const float* w_cr     = (const float*)d_in[7];
  const float* W0_rc    = (const float*)d_in[8];
  const float* b0_rc    = (const float*)d_in[9];
  const float* W0_cr    = (const float*)d_in[10];
  const float* b0_cr    = (const float*)d_in[11];
  const float* W1_rc    = (const float*)d_in[12];
  const float* b1_rc    = (const float*)d_in[13];
  const float* W1_cr    = (const float*)d_in[14];
  const float* b1_cr    = (const float*)d_in[15];

  const int DIN = 64;
  const int E  = in_sizes[2];
  const int NR = in_sizes[0] / DIN;
  const int NC = in_sizes[1] / DIN;
  const int NMAX = NR > NC ? NR : NC;

  float* ws = (float*)d_ws;
  float* invOutRow = ws;  ws += NR;
  float* invInCol  = ws;  ws += NC;
  float* invOutCol = ws;  ws += NC;
  float* invInRow  = ws;  ws += NR;
  float* agg   = ws;  ws += (long)NMAX * DH;
  float* hcol1 = ws;  ws += (long)NC * DH;
  float* hrow1 = ws;  ws += (long)NR * DH;
  float* hcol2 = ws;  ws += (long)NC * DH;
  float* hrow2 = ws;  ws += (long)NR * DH;

  // 1) degrees -> rsqrt(max(deg,1))
  const long ndeg = 2L * NR + 2L * NC;
  fill_zero<<<1024, 256, 0, stream>>>(invOutRow, ndeg);
  degree_kernel<<<(E + 255) / 256, 256, 0, stream>>>(src_rc, dst_rc, src_cr, dst_cr,
                                                     invOutRow, invInCol, invOutCol,
                                                     invInRow, E);
  rsqrt_clamp_kernel<<<(int)((ndeg + 255) / 256), 256, 0, stream>>>(invOutRow, (int)ndeg);

  // 2) one gconv = zero agg, fused scatter, WMMA gemm(+norm+bias+relu)
  auto gconv = [&](const float* h, const int* s, const int* d, const float* w,
                   const float* invO, const float* invI, const float* Wm,
                   const float* bm, float* outb, int ndst, int K) {
    fill_zero<<<2048, 256, 0, stream>>>(agg, (long)ndst * K);
    long tthreads = (long)E * (K >> 2);
    scatter_kernel<<<(int)((tthreads + 255) / 256), 256, 0, stream>>>(h, s, d, w, invO,
                                                                      agg, E, K);
    size_t shmem = (size_t)K * DH * sizeof(float);
    if (K == 64) {
      gemm_norm_bias_relu<64><<<(ndst + 15) / 16, 256, shmem, stream>>>(agg, invI, Wm, bm,
                                                                        outb, ndst);
    } else {
      gemm_norm_bias_relu<128><<<(ndst + 15) / 16, 256, shmem, stream>>>(agg, invI, Wm, bm,
                                                                         outb, ndst);
    }
  };

  // Layer 1
  gconv(feat_row, src_rc, dst_rc, w_rc, invOutRow, invInCol, W0_rc, b0_rc, hcol1, NC, DIN);
  gconv(feat_col, src_cr, dst_cr, w_cr, invOutCol, invInRow, W0_cr, b0_cr, hrow1, NR, DIN);
  // Layer 2
  gconv(hrow1, src_rc, dst_rc, w_rc, invOutRow, invInCol, W1_rc, b1_rc, hcol2, NC, DH);
  gconv(hcol1, src_cr, dst_cr, w_cr, invOutCol, invInRow, W1_cr, b1_cr, hrow2, NR, DH);

  // 3) readout
  colmean_kernel<<<DH, 256, 0, stream>>>(hrow2, hcol2, (float*)d_out, NR, NC);
}